// NestedLoop_29626684408030
// MI455X (gfx1250) — compile-verified
//
#include <hip/hip_runtime.h>

// out = inp * prod_{i=0..7}(1 + 28*i)   (elementwise, fp32, 64Mi elements)
// HBM-bound streaming kernel. Main kernel is branch-free (exact-cover grid):
// 8 x global_load_b128 NT in flight per thread, 7 exact-constant multiplies
// per element, 8 x global_store_b128 NT. Partial tiles / tail handled by
// separate guarded kernels (not launched for the 64Mi reference size).

typedef float v4f __attribute__((ext_vector_type(4)));

constexpr int kThreads = 256;   // 8 wave32 waves per block
constexpr int kUnroll  = 8;     // 8 x b128 loads outstanding per thread
constexpr int kTile    = kThreads * kUnroll;   // 2048 float4s per block

__device__ __forceinline__ v4f scale_chain(v4f v) {
    // Outer loop i=1..7: a *= (1 + 28*i). Factors 29,57,85,113,141,169,197
    // are exact in fp32; sequential multiplies mirror the reference's
    // outer-iteration rounding order.
#pragma unroll
    for (int i = 1; i < 8; ++i) {
        v *= (float)(1 + 28 * i);
    }
    return v;
}

// Full tiles only — no bounds checks, no exec manipulation.
__global__ __launch_bounds__(kThreads)
void nested_loop_scale_full(const v4f* __restrict__ in,
                            v4f* __restrict__ out) {
    const int base = blockIdx.x * kTile + threadIdx.x;

    v4f v[kUnroll];
#pragma unroll
    for (int u = 0; u < kUnroll; ++u) {
        v[u] = __builtin_nontemporal_load(&in[base + u * kThreads]);
    }
#pragma unroll
    for (int u = 0; u < kUnroll; ++u) {
        v[u] = scale_chain(v[u]);
    }
#pragma unroll
    for (int u = 0; u < kUnroll; ++u) {
        __builtin_nontemporal_store(v[u], &out[base + u * kThreads]);
    }
}

// Guarded float4 kernel for a partial last tile: covers [start4, n4).
__global__ __launch_bounds__(kThreads)
void nested_loop_scale_rem(const v4f* __restrict__ in,
                           v4f* __restrict__ out,
                           int start4, int n4) {
    const int k = start4 + blockIdx.x * kThreads + threadIdx.x;
    if (k < n4) {
        v4f v = __builtin_nontemporal_load(&in[k]);
        __builtin_nontemporal_store(scale_chain(v), &out[k]);
    }
}

// Scalar tail for (n % 4) elements.
__global__ __launch_bounds__(64)
void nested_loop_scale_tail(const float* __restrict__ in,
                            float* __restrict__ out,
                            int start, int n) {
    const int k = start + blockIdx.x * blockDim.x + threadIdx.x;
    if (k < n) {
        float a = in[k];
#pragma unroll
        for (int i = 1; i < 8; ++i) {
            a *= (float)(1 + 28 * i);
        }
        out[k] = a;
    }
}

extern "C" void kernel_launch(void* const* d_in, const int* in_sizes, int n_in,
                              void* d_out, int out_size, void* d_ws, size_t ws_size,
                              hipStream_t stream) {
    (void)in_sizes; (void)n_in; (void)d_ws; (void)ws_size;

    const float* in  = (const float*)d_in[0];
    float*       out = (float*)d_out;

    const int n          = out_size;   // 67,108,864 fp32 elements
    const int n4         = n / 4;      // 16,777,216 float4s
    const int full_tiles = n4 / kTile; // 8192 blocks (exact cover)
    const int rem4_start = full_tiles * kTile;

    if (full_tiles > 0) {
        nested_loop_scale_full<<<full_tiles, kThreads, 0, stream>>>(
            (const v4f*)in, (v4f*)out);
    }

    const int rem4 = n4 - rem4_start;          // leftover float4s (< kTile)
    if (rem4 > 0) {
        nested_loop_scale_rem<<<(rem4 + kThreads - 1) / kThreads, kThreads, 0, stream>>>(
            (const v4f*)in, (v4f*)out, rem4_start, n4);
    }

    const int tail = n - n4 * 4;               // leftover scalars (< 4)
    if (tail > 0) {
        nested_loop_scale_tail<<<1, 64, 0, stream>>>(in, out, n4 * 4, n);
    }
}